// AttnDecoderRNN_48292612276424
// MI455X (gfx1250) — compile-verified
//
#include <hip/hip_runtime.h>
#include <math.h>

// Shapes
#define Bsz 128
#define Ssz 400
#define Hsz 256
#define Esz 128
#define Vsz 50000
#define PADV 250
#define VP (Vsz + PADV)   // 50250

typedef __attribute__((ext_vector_type(2))) float v2f;
typedef __attribute__((ext_vector_type(8))) float v8f;

// ---- output layout (float elements), return order: h_new, p_final, p_gen, p_vocab, att_dist
#define OFF_HNEW 0
#define OFF_PFIN (Bsz*Hsz)
#define OFF_PGEN (OFF_PFIN + Bsz*VP)
#define OFF_PVOC (OFF_PGEN + Bsz)
#define OFF_ATT  (OFF_PVOC + (size_t)Bsz*Vsz)

// ---- workspace layout (float elements)
#define WS_X  0                      // B*E   (embedded tokens)
#define WS_DC (WS_X + Bsz*Esz)       // B*2H  (dec_ctx = [h_new, context])
#define WS_T  (WS_DC + Bsz*2*Hsz)    // B*E   (hidden->emb projection)

// ============================================================================
// K1: embedding gather + GRU cell. One block per batch row, 256 threads (H).
// ============================================================================
__global__ void k1_gru(const int* __restrict__ tok,
                       const float* __restrict__ hprev,
                       const float* __restrict__ emb,
                       const float* __restrict__ wih,
                       const float* __restrict__ whh,
                       const float* __restrict__ bih,
                       const float* __restrict__ bhh,
                       float* __restrict__ out,
                       float* __restrict__ ws) {
  const int b = blockIdx.x, tid = threadIdx.x;
  __shared__ float xs[Esz];
  __shared__ float hs[Hsz];
  const int t = tok[b];
  if (tid < Esz) xs[tid] = emb[(size_t)t * Esz + tid];
  hs[tid] = hprev[b * Hsz + tid];
  __syncthreads();

  float ir = bih[tid], iz = bih[Hsz + tid], inn = bih[2 * Hsz + tid];
  const float* wr = wih + (size_t)tid * Esz;
  const float* wz = wih + (size_t)(Hsz + tid) * Esz;
  const float* wn = wih + (size_t)(2 * Hsz + tid) * Esz;
  #pragma unroll 4
  for (int e = 0; e < Esz; ++e) {
    const float xv = xs[e];
    ir += xv * wr[e]; iz += xv * wz[e]; inn += xv * wn[e];
  }
  float hr = bhh[tid], hz = bhh[Hsz + tid], hn = bhh[2 * Hsz + tid];
  const float* vr = whh + (size_t)tid * Hsz;
  const float* vz = whh + (size_t)(Hsz + tid) * Hsz;
  const float* vn = whh + (size_t)(2 * Hsz + tid) * Hsz;
  #pragma unroll 4
  for (int h = 0; h < Hsz; ++h) {
    const float hv = hs[h];
    hr += hv * vr[h]; hz += hv * vz[h]; hn += hv * vn[h];
  }
  const float r = 1.f / (1.f + __expf(-(ir + hr)));
  const float z = 1.f / (1.f + __expf(-(iz + hz)));
  const float n = tanhf(inn + r * hn);
  const float hnew = (1.f - z) * n + z * hs[tid];

  out[OFF_HNEW + b * Hsz + tid] = hnew;
  ws[WS_DC + b * 2 * Hsz + tid] = hnew;             // dec_ctx[:, 0:H]
  if (tid < Esz) ws[WS_X + b * Esz + tid] = xs[tid];
}

// ============================================================================
// K2: fused attention: scores -> softmax -> context. One block per batch row.
// 8 waves; each wave owns one s-position at a time, wave32 shfl reduction.
// ============================================================================
__global__ void k2_attn(const float* __restrict__ enc,
                        const float* __restrict__ w_h,
                        const float* __restrict__ w_s,
                        const float* __restrict__ att_bias,
                        const float* __restrict__ attn_v,
                        float* __restrict__ out,
                        float* __restrict__ ws) {
  const int b = blockIdx.x, tid = threadIdx.x;
  __shared__ float sc[Ssz];
  __shared__ float whs[Hsz], cbs[Hsz], avs[Hsz];
  __shared__ float red[256];

  const float hn = out[OFF_HNEW + b * Hsz + tid];
  whs[tid] = w_h[tid];
  cbs[tid] = w_s[tid] * hn + att_bias[0];
  avs[tid] = attn_v[tid];
  __syncthreads();

  const int wave = tid >> 5, lane = tid & 31;
  const float* encb = enc + (size_t)b * Ssz * Hsz;

  for (int s = wave; s < Ssz; s += 8) {
    const float* er = encb + (size_t)s * Hsz;
    float acc = 0.f;
    #pragma unroll
    for (int h = lane; h < Hsz; h += 32)
      acc += tanhf(whs[h] * er[h] + cbs[h]) * avs[h];
    #pragma unroll
    for (int off = 16; off > 0; off >>= 1) acc += __shfl_xor(acc, off);
    if (lane == 0) sc[s] = acc;
  }
  __syncthreads();

  // softmax over S=400
  float m = -3.4e38f;
  for (int i = tid; i < Ssz; i += 256) m = fmaxf(m, sc[i]);
  red[tid] = m; __syncthreads();
  for (int off = 128; off > 0; off >>= 1) {
    if (tid < off) red[tid] = fmaxf(red[tid], red[tid + off]);
    __syncthreads();
  }
  m = red[0]; __syncthreads();
  float sum = 0.f;
  for (int i = tid; i < Ssz; i += 256) { const float e = __expf(sc[i] - m); sc[i] = e; sum += e; }
  red[tid] = sum; __syncthreads();
  for (int off = 128; off > 0; off >>= 1) {
    if (tid < off) red[tid] += red[tid + off];
    __syncthreads();
  }
  const float inv = 1.f / red[0];
  __syncthreads();
  for (int i = tid; i < Ssz; i += 256) {
    const float p = sc[i] * inv;
    sc[i] = p;
    out[OFF_ATT + (size_t)b * Ssz + i] = p;
  }
  __syncthreads();

  // context[h] = sum_s p[s] * enc[b,s,h]  (coalesced across tid)
  float acc = 0.f;
  for (int s = 0; s < Ssz; ++s) acc += sc[s] * encb[(size_t)s * Hsz + tid];
  ws[WS_DC + b * 2 * Hsz + Hsz + tid] = acc;        // dec_ctx[:, H:2H]
}

// ============================================================================
// K3: t = dec_ctx @ outh_w.T + outh_b   and   p_gen. One block per row, 128thr.
// ============================================================================
__global__ void k3_proj(const float* __restrict__ outh_w,
                        const float* __restrict__ outh_b,
                        const float* __restrict__ gen_w,
                        const float* __restrict__ gen_b,
                        float* __restrict__ out,
                        float* __restrict__ ws) {
  const int b = blockIdx.x, tid = threadIdx.x;     // 128 threads
  __shared__ float dc[2 * Hsz + Esz];              // 640
  __shared__ float red[128];
  for (int i = tid; i < 2 * Hsz; i += 128) dc[i] = ws[WS_DC + b * 2 * Hsz + i];
  dc[2 * Hsz + tid] = ws[WS_X + b * Esz + tid];
  __syncthreads();

  float acc = outh_b[tid];
  const float* wrow = outh_w + (size_t)tid * 2 * Hsz;
  #pragma unroll 4
  for (int k = 0; k < 2 * Hsz; ++k) acc += dc[k] * wrow[k];
  ws[WS_T + b * Esz + tid] = acc;

  float pg = 0.f;
  #pragma unroll
  for (int i = tid; i < 2 * Hsz + Esz; i += 128) pg += dc[i] * gen_w[i];
  red[tid] = pg; __syncthreads();
  for (int off = 64; off > 0; off >>= 1) {
    if (tid < off) red[tid] += red[tid + off];
    __syncthreads();
  }
  if (tid == 0) out[OFF_PGEN + b] = 1.f / (1.f + __expf(-(red[0] + gen_b[0])));
}

// ============================================================================
// K4: logits = t(128x128) @ outv_w^T(128x50000) + outv_b, via WMMA f32 16x16x4.
// Grid = 3125 N-tiles; 8 waves/block, each wave does one 16x16 M-tile, K=128.
// Logits land in the p_final slab (overwritten by K5 after softmax).
// ============================================================================
__global__ void k4_gemm(const float* __restrict__ t,
                        const float* __restrict__ outv_w,
                        const float* __restrict__ outv_b,
                        float* __restrict__ out) {
  const int tid = threadIdx.x;
  const int n_base = blockIdx.x * 16;
  __shared__ float bs[16][132];   // pad 128->132: bank = (4n + k) % 64, conflict-free
  __shared__ float bb[16];
  for (int i = tid; i < 16 * 128; i += 256) {
    const int r = i >> 7, c = i & 127;
    bs[r][c] = outv_w[(size_t)(n_base + r) * Esz + c];
  }
  if (tid < 16) bb[tid] = outv_b[n_base + tid];
  __syncthreads();

  const int wave = tid >> 5, lane = tid & 31;
  const int m_base = wave * 16;
  const int koff = (lane >> 4) * 2;               // ISA f32 16x4 A layout: lo lanes K=0/1, hi lanes K=2/3
  const int ncol = lane & 15;
  const float* arow = t + (size_t)(m_base + (lane & 15)) * Esz;

  float afrag[64];
  #pragma unroll
  for (int kk = 0; kk < 32; ++kk) {
    afrag[2 * kk]     = arow[4 * kk + koff];
    afrag[2 * kk + 1] = arow[4 * kk + koff + 1];
  }

  v8f c = {};
  #pragma unroll
  for (int kk = 0; kk < 32; ++kk) {
    v2f a;  a.x  = afrag[2 * kk];          a.y  = afrag[2 * kk + 1];
    v2f bv; bv.x = bs[ncol][4 * kk + koff]; bv.y = bs[ncol][4 * kk + koff + 1];
    c = __builtin_amdgcn_wmma_f32_16x16x4_f32(false, a, false, bv,
                                              (short)0, c, false, false);
  }

  float* pf = out + OFF_PFIN;
  const int mo = (lane >> 4) * 8;                 // D layout: vgpr i -> M=i (lo lanes) / M=8+i (hi)
  const float bias = bb[ncol];
  #pragma unroll
  for (int i = 0; i < 8; ++i) {
    const int row = m_base + mo + i;
    pf[(size_t)row * VP + n_base + ncol] = c[i] + bias;
  }
}

// ============================================================================
// K5: per-row softmax over V, p_vocab/p_final/PAD, then pointer scatter.
// One block per batch row, 256 threads.
// ============================================================================
__global__ void k5_final(const int* __restrict__ fiv,
                         float* __restrict__ out) {
  const int b = blockIdx.x, tid = threadIdx.x;
  __shared__ float red[256];
  float* pf = out + OFF_PFIN + (size_t)b * VP;     // currently holds logits
  float* pv = out + OFF_PVOC + (size_t)b * Vsz;
  const float pg = out[OFF_PGEN + b];

  float m = -3.4e38f;
  for (int j = tid; j < Vsz; j += 256) m = fmaxf(m, pf[j]);
  red[tid] = m; __syncthreads();
  for (int off = 128; off > 0; off >>= 1) {
    if (tid < off) red[tid] = fmaxf(red[tid], red[tid + off]);
    __syncthreads();
  }
  m = red[0]; __syncthreads();

  float sum = 0.f;
  for (int j = tid; j < Vsz; j += 256) { const float e = __expf(pf[j] - m); pv[j] = e; sum += e; }
  red[tid] = sum; __syncthreads();
  for (int off = 128; off > 0; off >>= 1) {
    if (tid < off) red[tid] += red[tid + off];
    __syncthreads();
  }
  const float inv = 1.f / red[0];
  __syncthreads();

  for (int j = tid; j < Vsz; j += 256) {
    const float p = pv[j] * inv;
    pv[j] = p;                 // p_vocab
    pf[j] = pg * p;            // p_final base
  }
  for (int j = Vsz + tid; j < VP; j += 256) pf[j] = 0.f;   // PAD region
  __syncthreads();

  const float* att = out + OFF_ATT + (size_t)b * Ssz;
  const float og = 1.f - pg;
  for (int s = tid; s < Ssz; s += 256)
    atomicAdd(&pf[fiv[b * Ssz + s]], og * att[s]);          // global_atomic_add_f32
}

// ============================================================================
extern "C" void kernel_launch(void* const* d_in, const int* in_sizes, int n_in,
                              void* d_out, int out_size, void* d_ws, size_t ws_size,
                              hipStream_t stream) {
  const int*   tok      = (const int*)  d_in[0];
  const float* hprev    = (const float*)d_in[1];
  const float* enc      = (const float*)d_in[2];
  const int*   fiv      = (const int*)  d_in[3];
  const float* emb      = (const float*)d_in[4];
  const float* wih      = (const float*)d_in[5];
  const float* whh      = (const float*)d_in[6];
  const float* bih      = (const float*)d_in[7];
  const float* bhh      = (const float*)d_in[8];
  const float* w_h      = (const float*)d_in[9];
  const float* w_s      = (const float*)d_in[10];
  const float* att_bias = (const float*)d_in[11];
  const float* attn_v   = (const float*)d_in[12];
  const float* gen_w    = (const float*)d_in[13];
  const float* gen_b    = (const float*)d_in[14];
  const float* outh_w   = (const float*)d_in[15];
  const float* outh_b   = (const float*)d_in[16];
  const float* outv_w   = (const float*)d_in[17];
  const float* outv_b   = (const float*)d_in[18];

  float* out = (float*)d_out;
  float* ws  = (float*)d_ws;

  k1_gru <<<Bsz,        256, 0, stream>>>(tok, hprev, emb, wih, whh, bih, bhh, out, ws);
  k2_attn<<<Bsz,        256, 0, stream>>>(enc, w_h, w_s, att_bias, attn_v, out, ws);
  k3_proj<<<Bsz,        128, 0, stream>>>(outh_w, outh_b, gen_w, gen_b, out, ws);
  k4_gemm<<<Vsz / 16,   256, 0, stream>>>(ws + WS_T, outv_w, outv_b, out);
  k5_final<<<Bsz,       256, 0, stream>>>(fiv, out);
}